// ChannelGate_45896020525450
// MI455X (gfx1250) — compile-verified
//
#include <hip/hip_runtime.h>
#include <math.h>

// Problem constants (from reference)
#define Bn    8
#define Cn    64
#define Fn    3
#define Hh    192
#define Wn    192
#define OH    182          // 192 - 11 + 1
#define OW    182
#define WSZ   11
#define SIGMA 1.5f
#define C1    0.01f        // (0.01*10)^2
#define C2    0.09f        // (0.03*10)^2

#define STRIP 16
#define MAXR  (STRIP + WSZ - 1)   // 26 input rows per strip
#define KPAD  28                  // K dim padded to multiple of 4 for WMMA
#define PITCH 184                 // padded LDS row pitch
#define NT    12                  // ceil(182/16) column tiles per strip

typedef __attribute__((ext_vector_type(2))) float v2f;
typedef __attribute__((ext_vector_type(8))) float v8f;

__device__ __forceinline__ v8f wmma4(v2f a, v2f b, v8f c) {
    return __builtin_amdgcn_wmma_f32_16x16x4_f32(false, a, false, b, (short)0, c,
                                                 false, false);
}

__device__ __forceinline__ void make_window(float* w) {
    float s = 0.f;
#pragma unroll
    for (int i = 0; i < WSZ; ++i) {
        float d = (float)(i - WSZ / 2);
        w[i] = __expf(-(d * d) / (2.f * SIGMA * SIGMA));
        s += w[i];
    }
    float inv = 1.f / s;
#pragma unroll
    for (int i = 0; i < WSZ; ++i) w[i] *= inv;
}

// ---------------------------------------------------------------------------
// Kernel A: per (b,f) image -> mu_f, sig_f fields (separable 11x11 blur)
// grid = Bn*Fn blocks, 256 threads
// ---------------------------------------------------------------------------
__global__ __launch_bounds__(256) void moments_f_kernel(
        const float* __restrict__ xf, float* __restrict__ muF, float* __restrict__ sigF) {
    __shared__ float Hf [MAXR * PITCH];
    __shared__ float Hff[MAXR * PITCH];
    float w[WSZ];
    make_window(w);
    const int img = blockIdx.x;                          // b*Fn + f
    const float* src = xf + (size_t)img * Hh * Wn;
    float* muO = muF + (size_t)img * OH * OW;
    float* sgO = sigF + (size_t)img * OH * OW;
    const int tid = threadIdx.x;

    for (int y0 = 0; y0 < OH; y0 += STRIP) {
        const int nOut = min(STRIP, OH - y0);
        const int nIn  = nOut + WSZ - 1;
        const bool pf  = (y0 + STRIP) < OH;              // uniform
        for (int idx = tid; idx < nIn * OW; idx += 256) {
            int r = idx / OW, xc = idx - r * OW;
            const float* row = src + (y0 + r) * Wn + xc;
            if (pf) __builtin_prefetch(row + STRIP * Wn, 0, 3);  // next strip
            float a = 0.f, a2 = 0.f;
#pragma unroll
            for (int k = 0; k < WSZ; ++k) { float v = row[k]; a += w[k] * v; a2 += w[k] * v * v; }
            Hf [r * PITCH + xc] = a;
            Hff[r * PITCH + xc] = a2;
        }
        __syncthreads();
        for (int idx = tid; idx < nOut * OW; idx += 256) {
            int r = idx / OW, xc = idx - r * OW;
            float mu = 0.f, e2 = 0.f;
#pragma unroll
            for (int j = 0; j < WSZ; ++j) {
                mu += w[j] * Hf [(r + j) * PITCH + xc];
                e2 += w[j] * Hff[(r + j) * PITCH + xc];
            }
            muO[(y0 + r) * OW + xc] = mu;
            sgO[(y0 + r) * OW + xc] = e2 - mu * mu;
        }
        __syncthreads();
    }
}

// ---------------------------------------------------------------------------
// Kernel B: per (b,c) pair. Horizontal 11-tap passes on VALU into LDS strip
// buffers; vertical 11-tap pass as banded matmul on the matrix pipe:
//   Out[16x16] = Band[16x28] x H[28x16]   via 7x V_WMMA_F32_16X16X4_F32
// per signal (mu_x, E[x^2], E[f*x]). SSIM formula + reduction on VALU.
// Next-strip inputs prefetched (global_prefetch_b8, speculative -> edge-safe).
// grid = Bn*Cn blocks, 256 threads (8 waves). LDS ~62.8 KB.
// ---------------------------------------------------------------------------
__global__ __launch_bounds__(256) void ssim_kernel(
        const float* __restrict__ x, const float* __restrict__ xf,
        const float* __restrict__ muF, const float* __restrict__ sigF,
        float* __restrict__ ssim_out /* [B][F][C] */) {
    __shared__ float Hx [KPAD * PITCH];
    __shared__ float Hxx[KPAD * PITCH];
    __shared__ float Hp [KPAD * PITCH];
    __shared__ float red[256];
    float w[WSZ];
    make_window(w);
    const int blk = blockIdx.x;
    const int b = blk >> 6, c = blk & 63;
    const float* xs = x + (size_t)(b * Cn + c) * Hh * Wn;
    const int tid  = threadIdx.x;
    const int lane = tid & 31;
    const int wv   = tid >> 5;               // wave id 0..7
    const int m    = lane & 15;              // row (A) / col-in-tile (B,D)
    const int koff = (lane >> 4) << 1;       // K offset for lane halves
    const int rbase = (lane < 16) ? 0 : 8;   // D-tile row base for this lane

    // one-time zero of LDS buffers (padding rows/cols stay 0 forever -> no
    // NaN poison can enter the WMMA band columns)
    for (int i = tid; i < KPAD * PITCH; i += 256) { Hx[i] = 0.f; Hxx[i] = 0.f; Hp[i] = 0.f; }
    __syncthreads();

    float a0 = 0.f, a1 = 0.f, a2 = 0.f;

    for (int y0 = 0; y0 < OH; y0 += STRIP) {
        const int nOut = min(STRIP, OH - y0);
        const int nIn  = nOut + WSZ - 1;
        const bool pf  = (y0 + STRIP) < OH;              // uniform

        // build banded A-matrix chunks for the vertical blur (registers only)
        v2f aB[7];
#pragma unroll
        for (int kc = 0; kc < 7; ++kc) {
            int k0 = kc * 4 + koff;
            float v0 = 0.f, v1 = 0.f;
            if (m < nOut) {
#pragma unroll
                for (int j = 0; j < WSZ; ++j) {
                    if (k0     == m + j) v0 = w[j];
                    if (k0 + 1 == m + j) v1 = w[j];
                }
            }
            aB[kc].x = v0; aB[kc].y = v1;
        }

        // horizontal pass for x moments (+ prefetch next strip of x)
        for (int idx = tid; idx < nIn * OW; idx += 256) {
            int r = idx / OW, xc = idx - r * OW;
            const float* row = xs + (y0 + r) * Wn + xc;
            if (pf) __builtin_prefetch(row + STRIP * Wn, 0, 3);
            float hx = 0.f, hxx = 0.f;
#pragma unroll
            for (int k = 0; k < WSZ; ++k) { float v = row[k]; hx += w[k] * v; hxx += w[k] * v * v; }
            Hx [r * PITCH + xc] = hx;
            Hxx[r * PITCH + xc] = hxx;
        }
        __syncthreads();

        for (int f = 0; f < Fn; ++f) {
            const float* fs = xf + (size_t)(b * Fn + f) * Hh * Wn;
            // horizontal pass for cross product xf*x (+ prefetch next strip)
            for (int idx = tid; idx < nIn * OW; idx += 256) {
                int r = idx / OW, xc = idx - r * OW;
                const float* rowx = xs + (y0 + r) * Wn + xc;
                const float* rowf = fs + (y0 + r) * Wn + xc;
                if (pf) __builtin_prefetch(rowf + STRIP * Wn, 0, 3);
                float hp = 0.f;
#pragma unroll
                for (int k = 0; k < WSZ; ++k) hp += w[k] * rowf[k] * rowx[k];
                Hp[r * PITCH + xc] = hp;
            }
            __syncthreads();

            // vertical pass on the matrix pipe + SSIM epilogue
            const float* muFp = muF + (size_t)(b * Fn + f) * OH * OW;
            const float* sgFp = sigF + (size_t)(b * Fn + f) * OH * OW;
            float sum = 0.f;
            for (int t = wv; t < NT; t += 8) {           // wave-uniform tiles
                const int cb = t * 16;                   // tile column base
                v8f mAcc = {}, xAcc = {}, pAcc = {};
#pragma unroll
                for (int kc = 0; kc < 7; ++kc) {
                    int k = kc * 4 + koff;
                    int o0 = k * PITCH + cb + m;
                    v2f bx; bx.x = Hx [o0]; bx.y = Hx [o0 + PITCH];
                    v2f b2; b2.x = Hxx[o0]; b2.y = Hxx[o0 + PITCH];
                    v2f bp; bp.x = Hp [o0]; bp.y = Hp [o0 + PITCH];
                    mAcc = wmma4(aB[kc], bx, mAcc);
                    xAcc = wmma4(aB[kc], b2, xAcc);
                    pAcc = wmma4(aB[kc], bp, pAcc);
                }
                const int col = cb + m;
                if (col < OW) {
#pragma unroll
                    for (int i = 0; i < 8; ++i) {
                        int r = i + rbase;
                        if (r < nOut) {
                            float mux = mAcc[i], exx = xAcc[i], ep = pAcc[i];
                            int go = (y0 + r) * OW + col;
                            float muf = muFp[go], sgf = sgFp[go];
                            float sgx  = exx - mux * mux;
                            float sgfx = ep - muf * mux;
                            float v1 = 2.f * sgfx + C2;
                            float v2 = sgf + sgx + C2;
                            float num = (2.f * muf * mux + C1) * v1;
                            float den = (muf * muf + mux * mux + C1) * v2;
                            sum += num / den;
                        }
                    }
                }
            }
            if      (f == 0) a0 += sum;
            else if (f == 1) a1 += sum;
            else             a2 += sum;
            __syncthreads();
        }
    }

    // block reduction of the three per-f sums
    const float inv = 1.f / (float)(OH * OW);
    for (int f = 0; f < Fn; ++f) {
        red[tid] = (f == 0) ? a0 : (f == 1) ? a1 : a2;
        __syncthreads();
        for (int s = 128; s > 0; s >>= 1) {
            if (tid < s) red[tid] += red[tid + s];
            __syncthreads();
        }
        if (tid == 0) ssim_out[(b * Fn + f) * Cn + c] = red[0] * inv;
        __syncthreads();
    }
}

// ---------------------------------------------------------------------------
// Kernel C: spatial gate conv + 2-layer MLP via V_WMMA_F32_16X16X4_F32
// 1 block, 128 threads (4 waves); wave t owns N-tile t (16 cols of 64).
// ---------------------------------------------------------------------------
__global__ __launch_bounds__(128) void gate_mlp_kernel(
        const float* __restrict__ ssim, const float* __restrict__ sw,
        const float* __restrict__ W1, const float* __restrict__ b1,
        const float* __restrict__ W2, const float* __restrict__ b2,
        float* __restrict__ out) {
    __shared__ float Ag[16 * 64];
    __shared__ float Hm[16 * 64];
    const int tid = threadIdx.x;

    // gate: Conv2d(3,1,(3,1),pad=(1,0)) over the C axis, then ReLU
    for (int idx = tid; idx < 16 * 64; idx += 128) {
        int b = idx >> 6, c = idx & 63;
        float v = 0.f;
        if (b < Bn) {
            float s = 0.f;
#pragma unroll
            for (int f = 0; f < Fn; ++f) {
                const float* sp = ssim + (b * Fn + f) * Cn;
                float l = (c > 0)  ? sp[c - 1] : 0.f;
                float mm = sp[c];
                float r = (c < 63) ? sp[c + 1] : 0.f;
                s += sw[f * 3 + 0] * l + sw[f * 3 + 1] * mm + sw[f * 3 + 2] * r;
            }
            v = fmaxf(s, 0.f);
        }
        Ag[idx] = v;
    }
    __syncthreads();

    const int lane = tid & 31;
    const int wv   = tid >> 5;               // 0..3 -> N tile
    const int m    = lane & 15;
    const int koff = (lane >> 4) << 1;
    const int n    = wv * 16 + m;

    // GEMM1: h1 = relu(Ag @ W1^T + b1)
    v8f acc = {};
    for (int kc = 0; kc < 16; ++kc) {
        int k = kc * 4 + koff;
        v2f av; av.x = Ag[m * 64 + k]; av.y = Ag[m * 64 + k + 1];
        v2f bv; bv.x = W1[n * 64 + k]; bv.y = W1[n * 64 + k + 1];
        acc = wmma4(av, bv, acc);
    }
    {
        float bias = b1[n];
        int rb = (lane < 16) ? 0 : 8;
#pragma unroll
        for (int i = 0; i < 8; ++i)
            Hm[(i + rb) * 64 + n] = fmaxf(acc[i] + bias, 0.f);
    }
    __syncthreads();

    // GEMM2: out = sigmoid(Hm @ W2^T + b2)
    v8f acc2 = {};
    for (int kc = 0; kc < 16; ++kc) {
        int k = kc * 4 + koff;
        v2f av; av.x = Hm[m * 64 + k]; av.y = Hm[m * 64 + k + 1];
        v2f bv; bv.x = W2[n * 64 + k]; bv.y = W2[n * 64 + k + 1];
        acc2 = wmma4(av, bv, acc2);
    }
    {
        float bias = b2[n];
        int rb = (lane < 16) ? 0 : 8;
#pragma unroll
        for (int i = 0; i < 8; ++i) {
            int row = i + rb;
            if (row < Bn) {
                float z = acc2[i] + bias;
                out[row * 64 + n] = 1.f / (1.f + __expf(-z));
            }
        }
    }
}

// ---------------------------------------------------------------------------
extern "C" void kernel_launch(void* const* d_in, const int* in_sizes, int n_in,
                              void* d_out, int out_size, void* d_ws, size_t ws_size,
                              hipStream_t stream) {
    const float* x  = (const float*)d_in[0];
    const float* xf = (const float*)d_in[1];
    const float* sw = (const float*)d_in[2];
    const float* W1 = (const float*)d_in[3];
    const float* b1 = (const float*)d_in[4];
    const float* W2 = (const float*)d_in[5];
    const float* b2 = (const float*)d_in[6];
    float* out = (float*)d_out;

    // workspace: mu_f and sig_f fields, 2 * 8*3*182*182 floats = 6.36 MB
    float* muF  = (float*)d_ws;
    float* sigF = muF + (size_t)Bn * Fn * OH * OW;

    // ssim_info is the second output segment; kernel C consumes it in-place
    float* ssim = out + Bn * Cn;

    moments_f_kernel<<<Bn * Fn, 256, 0, stream>>>(xf, muF, sigF);
    ssim_kernel<<<Bn * Cn, 256, 0, stream>>>(x, xf, muF, sigF, ssim);
    gate_mlp_kernel<<<1, 128, 0, stream>>>(ssim, sw, W1, b1, W2, b2, out);
}